// LengthRegulator_15006615732937
// MI455X (gfx1250) — compile-verified
//
#include <hip/hip_runtime.h>
#include <hip/hip_bf16.h>

// LengthRegulator for MI455X (gfx1250).
// Pure data-movement op: ~32MB read + ~150MB write -> HBM-bound (~8us floor
// at 23.3 TB/s). Strategy: inverse mapping (no searchsorted) — each input row
// knows its output slice from the duration cumsum. Global->LDS staging is done
// by the CDNA5 Tensor Data Mover (tensor_load_to_lds, TENSORcnt), double-
// buffered so the DMA of tile k+1 overlaps the replicated coalesced row
// stores of tile k.

#define LR_B 32
#define LR_T 512
#define LR_D 512
#define TILE_ROWS 16          // rows per TDM tile: 16*512*4B = 32KB LDS
#define TILES_PER_BLOCK 2     // double buffer: 64KB LDS per block

typedef __attribute__((ext_vector_type(4))) unsigned int v4u;
typedef __attribute__((ext_vector_type(8))) int          v8i;
typedef __attribute__((ext_vector_type(4))) int          v4i;

// ---------------------------------------------------------------------------
// Kernel 1: per-batch inclusive cumsum of durations (Hillis-Steele in LDS),
// plus pass-through copy of durations into the tail of d_out.
// ---------------------------------------------------------------------------
__global__ void lr_scan(const int* __restrict__ dur,
                        int* __restrict__ cum,
                        int* __restrict__ totals,
                        int* __restrict__ dur_out) {
  __shared__ int s[LR_T];
  const int b = blockIdx.x;
  const int tid = threadIdx.x;

  const int d = dur[b * LR_T + tid];
  s[tid] = d;
  __syncthreads();

#pragma unroll
  for (int off = 1; off < LR_T; off <<= 1) {
    const int a = s[tid];
    const int u = (tid >= off) ? s[tid - off] : 0;
    __syncthreads();
    s[tid] = a + u;
    __syncthreads();
  }

  cum[b * LR_T + tid] = s[tid];
  if (tid == LR_T - 1) totals[b] = s[LR_T - 1];
  dur_out[b * LR_T + tid] = d;  // tuple output #2 (bit-preserving)
}

// ---------------------------------------------------------------------------
// TDM descriptor build + issue: load a [rows x 512] f32 tile (row-contiguous,
// row stride 512) from global into LDS. D# layout per CDNA5 ISA ch. 8.
// 6-arg builtin on this toolchain:
//   (uint32x4 g0, int32x8 g1, int32x4 g2, int32x4 g3, int32x8 extra, i32 cpol)
// ---------------------------------------------------------------------------
__device__ __forceinline__ void tdm_load_tile_f32(const float* gsrc,
                                                  unsigned int lds_byte_addr,
                                                  int rows) {
  const unsigned long long ga = (unsigned long long)(uintptr_t)gsrc;

  v4u g0;
  g0.x = 1u;                                   // count=1 (valid), gather off
  g0.y = lds_byte_addr;                        // lds_addr
  g0.z = (unsigned int)ga;                     // global_addr[31:0]
  g0.w = ((unsigned int)(ga >> 32) & 0x01FFFFFFu) | (2u << 30); // [56:32] | type=2

  v8i g1;
  g1.s0 = (int)(2u << 16);                     // wg_mask=0 | data_size=2 (4B)
  g1.s1 = (int)((LR_D & 0xFFFF) << 16);        // tensor_dim0[15:0] (=512)
  g1.s2 = (int)((LR_T & 0xFFFF) << 16);        // dim0 hi=0 | tensor_dim1[15:0]
  g1.s3 = (int)((LR_D & 0xFFFF) << 16);        // dim1 hi=0 | tile_dim0=512
  g1.s4 = rows & 0xFFFF;                       // tile_dim1 | tile_dim2=0
  g1.s5 = LR_D;                                // tensor_dim0_stride[31:0]=512
  g1.s6 = 0;                                   // stride hi | dim1_stride lo
  g1.s7 = 0;                                   // dim1_stride hi (unused, 2D)

  v4i g2 = {0, 0, 0, 0};                       // 2D tensor: groups 2/3 unused
  v4i g3 = {0, 0, 0, 0};
  v8i gx = {0, 0, 0, 0, 0, 0, 0, 0};           // extra group (zero-filled)

  __builtin_amdgcn_tensor_load_to_lds(g0, g1, g2, g3, gx, 0);
}

// ---------------------------------------------------------------------------
// Fan one 16-row LDS tile out to the output: each wave owns 2 rows, caches
// the row in 16 VGPRs (4 x float4), stores each replica as a coalesced 2KB row.
// ---------------------------------------------------------------------------
__device__ __forceinline__ void fanout_tile(const float* __restrict__ tile,
                                            const int* __restrict__ cumb,
                                            float* __restrict__ out,
                                            int b, int t0, int wid, int lane,
                                            int max_len) {
#pragma unroll
  for (int j = 0; j < 2; ++j) {
    const int r = wid * 2 + j;          // local row 0..15
    const int t = t0 + r;
    const int end   = cumb[t];
    const int start = (t == 0) ? 0 : cumb[t - 1];

    const float* lrow = &tile[r * LR_D + lane * 4];
    const float4 f0 = *(const float4*)(lrow + 0);
    const float4 f1 = *(const float4*)(lrow + 128);
    const float4 f2 = *(const float4*)(lrow + 256);
    const float4 f3 = *(const float4*)(lrow + 384);

    for (int p = start; p < end; ++p) {
      float* o = out + ((size_t)b * max_len + p) * LR_D + lane * 4;
      *(float4*)(o + 0)   = f0;
      *(float4*)(o + 128) = f1;
      *(float4*)(o + 256) = f2;
      *(float4*)(o + 384) = f3;
    }
  }
}

// ---------------------------------------------------------------------------
// Kernel 2: expansion. Grid (B, T/32), 256 threads = 8 waves.
// Wave 0 issues both TDM tile loads back-to-back; s_wait_tensorcnt(1) lets
// the block fan out tile 0 while the TDM still streams tile 1 (TDM ops from
// one wave complete in order).
// ---------------------------------------------------------------------------
__global__ void __launch_bounds__(256)
lr_expand(const float* __restrict__ enc,
          const int* __restrict__ cum,
          float* __restrict__ out,
          int max_len) {
  __shared__ __align__(128) float tile[TILES_PER_BLOCK][TILE_ROWS * LR_D];

  const int b  = blockIdx.x;
  const int t0 = blockIdx.y * (TILE_ROWS * TILES_PER_BLOCK);
  const int tid = threadIdx.x;

  if (tid == 0) {
    tdm_load_tile_f32(enc + ((size_t)b * LR_T + t0) * LR_D,
                      (unsigned int)(uintptr_t)&tile[0][0], TILE_ROWS);
    tdm_load_tile_f32(enc + ((size_t)b * LR_T + t0 + TILE_ROWS) * LR_D,
                      (unsigned int)(uintptr_t)&tile[1][0], TILE_ROWS);
  }

  const int wid  = tid >> 5;   // wave 0..7
  const int lane = tid & 31;
  const int* cumb = cum + b * LR_T;

  // Tile 0 landed when wave 0's TENSORcnt drops to <=1 (in-order completion).
  __builtin_amdgcn_s_wait_tensorcnt(1);
  __syncthreads();
  fanout_tile(tile[0], cumb, out, b, t0, wid, lane, max_len);

  // Tile 1 landed when TENSORcnt reaches 0.
  __builtin_amdgcn_s_wait_tensorcnt(0);
  __syncthreads();
  fanout_tile(tile[1], cumb, out, b, t0 + TILE_ROWS, wid, lane, max_len);
}

// ---------------------------------------------------------------------------
// Kernel 3: zero the invalid tail rows p in [total[b], max_len).
// Grid (B, ceil(max_len/8)), 256 threads = 8 waves, one row per wave.
// ---------------------------------------------------------------------------
__global__ void __launch_bounds__(256)
lr_zerofill(const int* __restrict__ totals, float* __restrict__ out, int max_len) {
  const int b    = blockIdx.x;
  const int wid  = threadIdx.x >> 5;
  const int lane = threadIdx.x & 31;
  const int p = blockIdx.y * 8 + wid;
  const int total = totals[b];
  if (p >= max_len || p < total) return;

  const float4 z = {0.f, 0.f, 0.f, 0.f};
  float* o = out + ((size_t)b * max_len + p) * LR_D + lane * 4;
  *(float4*)(o + 0)   = z;
  *(float4*)(o + 128) = z;
  *(float4*)(o + 256) = z;
  *(float4*)(o + 384) = z;
}

// ---------------------------------------------------------------------------
extern "C" void kernel_launch(void* const* d_in, const int* in_sizes, int n_in,
                              void* d_out, int out_size, void* d_ws, size_t ws_size,
                              hipStream_t stream) {
  const float* enc = (const float*)d_in[0];   // (B, T, D) f32
  const int*   dur = (const int*)d_in[1];     // (B, T) i32
  float* out = (float*)d_out;

  // out_size = B*max_len*D + B*T  (tuple: expanded output + durations)
  const int max_len = (out_size - LR_B * LR_T) / (LR_B * LR_D);

  int* cum    = (int*)d_ws;                   // B*T ints
  int* totals = cum + LR_B * LR_T;            // B ints
  int* dur_out = (int*)(out + (size_t)LR_B * max_len * LR_D);

  lr_scan<<<LR_B, LR_T, 0, stream>>>(dur, cum, totals, dur_out);
  lr_expand<<<dim3(LR_B, LR_T / (TILE_ROWS * TILES_PER_BLOCK)), 256, 0, stream>>>(
      enc, cum, out, max_len);
  lr_zerofill<<<dim3(LR_B, (max_len + 7) / 8), 256, 0, stream>>>(totals, out, max_len);
}